// ExpertChoiceRouter_12567074308596
// MI455X (gfx1250) — compile-verified
//
#include <hip/hip_runtime.h>
#include <hip/hip_bf16.h>
#include <math.h>

typedef __attribute__((ext_vector_type(2))) float v2f;
typedef __attribute__((ext_vector_type(8))) float v8f;

// Problem shape (fixed by reference setup_inputs)
constexpr int B = 4, T = 4096, D = 2048;
constexpr int TOPK = T / 2;                 // CAPACITY = 0.5
constexpr float ALPHA = 0.1f;
constexpr float AUX_W = 0.001f;
constexpr float Z_W   = 0.001f;

// Flat output layout (reference return order)
constexpr int OUT_SEL = 0;                  // (B, k, 1) token indices as float
constexpr int OUT_GW  = B * TOPK;           // (B, k, 1) gate weights
constexpr int OUT_LOG = 2 * B * TOPK;       // (B, T, 1) raw logits
constexpr int OUT_AUX = OUT_LOG + B * T;    // scalar
constexpr int OUT_Z   = OUT_AUX + 1;        // scalar

// Workspace layout (floats)
constexpr int WS_GV  = 0;                   // B*T gate values
constexpr int WS_TGT = B * T;               // B*T BCE targets (top-k mask)

// ---------------------------------------------------------------------------
// Kernel 1: logits GEMV via V_WMMA_F32_16X16X4_F32.
// A = 16 tokens x 4 features (per ISA layout: lanes 0-15 carry K=0,1;
// lanes 16-31 carry K=2,3). B = W[d..d+3] broadcast to all 16 columns
// (lane-half h loads W[d+2h], W[d+2h+1]), so every accumulator column holds
// the 16 token logits. Memory-bound: streams x exactly once.
// ---------------------------------------------------------------------------
__global__ __launch_bounds__(256)
void router_logits_wmma(const float* __restrict__ x, const float* __restrict__ W,
                        float* __restrict__ out, float* __restrict__ gv)
{
    const int lane = threadIdx.x & 31;
    const int wave = threadIdx.x >> 5;
    const int tokenBase = blockIdx.x * 128 + wave * 16;   // flat over B*T
    const int b  = tokenBase / T;
    const int t0 = tokenBase - b * T;
    const int half = lane >> 4;                           // 0 or 1
    const int mrow = lane & 15;                           // token row within tile
    const int koff = half * 2;                            // K offset inside 4-chunk

    const float* __restrict__ xrow = x + (size_t)b * T * D + (size_t)(t0 + mrow) * D + koff;
    const float* __restrict__ wrow = W + koff;

    v8f c = {};
    for (int d0 = 0; d0 < D; d0 += 4) {
        v2f a  = *(const v2f*)(xrow + d0);
        v2f bb = *(const v2f*)(wrow + d0);
        c = __builtin_amdgcn_wmma_f32_16x16x4_f32(
                /*neg_a=*/false, a, /*neg_b=*/false, bb,
                /*c_mod=*/(short)0, c, /*reuse_a=*/false, /*reuse_b=*/false);
    }

    // C/D layout: lane n holds column n; VGPR r holds row r (lanes 0-15) or
    // row r+8 (lanes 16-31). All columns identical -> col 0 lanes write.
    if (mrow == 0) {
#pragma unroll
        for (int r = 0; r < 8; ++r) {
            const int m = r + half * 8;
            const float l = c[r];
            out[OUT_LOG + b * T + t0 + m] = l;                       // raw_logits
            gv[b * T + t0 + m] = ALPHA / (1.0f + __expf(-l));        // sigmoid*alpha
        }
    }
}

// ---------------------------------------------------------------------------
// Kernel 2: per-row top-k (k = T/2) with jax tie-breaking (lower index wins),
// emitting ascending-sorted indices + gathered gate weights + target mask.
// One 1024-thread block per batch row; each thread owns 4 contiguous tokens.
// ---------------------------------------------------------------------------
__global__ __launch_bounds__(1024)
void topk_select(const float* __restrict__ gv, float* __restrict__ out,
                 float* __restrict__ tgt)
{
    __shared__ unsigned scan[1024];
    __shared__ int scnt;

    const int b   = blockIdx.x;
    const int tid = threadIdx.x;
    const float* __restrict__ row = gv + b * T;

    unsigned mybits[4];
#pragma unroll
    for (int j = 0; j < 4; ++j)
        mybits[j] = __float_as_uint(row[tid * 4 + j]);   // positive floats: uint order == float order

    // Binary search for the largest threshold with count(>= thr) >= k.
    unsigned thr = 0u;
    for (int bit = 30; bit >= 0; --bit) {
        const unsigned cand = thr | (1u << bit);
        if (tid == 0) scnt = 0;
        __syncthreads();
        int local = 0;
#pragma unroll
        for (int j = 0; j < 4; ++j) local += (mybits[j] >= cand);
        if (local) atomicAdd(&scnt, local);
        __syncthreads();
        if (scnt >= TOPK) thr = cand;
        __syncthreads();
    }

    // Packed counts: low 16 bits = #(v > thr), high 16 bits = #(v == thr).
    unsigned packed = 0;
#pragma unroll
    for (int j = 0; j < 4; ++j) {
        packed += (mybits[j] >  thr) ? 1u       : 0u;
        packed += (mybits[j] == thr) ? 0x10000u : 0u;
    }
    scan[tid] = packed;
    __syncthreads();
    for (int off = 1; off < 1024; off <<= 1) {
        const unsigned v = (tid >= off) ? scan[tid - off] : 0u;
        __syncthreads();
        scan[tid] += v;
        __syncthreads();
    }
    const unsigned total = scan[1023];
    const unsigned excl  = scan[tid] - packed;   // exclusive prefix for this thread
    const unsigned cntGt = total & 0xFFFFu;
    const unsigned need  = (unsigned)TOPK - cntGt;   // equals to take (ascending index)
    unsigned gtp = excl & 0xFFFFu;
    unsigned eqp = excl >> 16;

#pragma unroll
    for (int j = 0; j < 4; ++j) {
        const int t = tid * 4 + j;
        const unsigned u = mybits[j];
        const bool gt = (u > thr);
        const bool eq = (u == thr);
        const bool sel = gt || (eq && eqp < need);
        if (sel) {
            const unsigned pos = gtp + (eqp < need ? eqp : need);  // ascending order
            out[OUT_SEL + b * TOPK + pos] = (float)t;              // selected_tokens
            out[OUT_GW  + b * TOPK + pos] = __uint_as_float(u);    // gate_weights
        }
        tgt[b * T + t] = sel ? 1.0f : 0.0f;
        gtp += gt ? 1u : 0u;
        eqp += eq ? 1u : 0u;
    }
}

// ---------------------------------------------------------------------------
// Kernel 3: aux BCE loss (sum reduction over B*T) + z-loss (per-row
// logsumexp). Single block, deterministic (no cross-block float atomics).
// ---------------------------------------------------------------------------
__global__ __launch_bounds__(1024)
void losses_kernel(float* __restrict__ out, const float* __restrict__ tgt)
{
    __shared__ float red[1024];
    const int tid = threadIdx.x;
    const float* __restrict__ logits = out + OUT_LOG;

    // Numerically-stable BCE-with-logits, reduction='sum'
    float acc = 0.0f;
    for (int i = tid; i < B * T; i += 1024) {
        const float l = logits[i];
        const float tg = tgt[i];
        acc += fmaxf(l, 0.0f) - l * tg + log1pf(__expf(-fabsf(l)));
    }
    red[tid] = acc;
    __syncthreads();
    for (int off = 512; off > 0; off >>= 1) {
        if (tid < off) red[tid] += red[tid + off];
        __syncthreads();
    }
    const float bce = red[0];
    __syncthreads();

    // z-loss: mean over rows of logsumexp(row)^2
    float zsq = 0.0f;
    for (int b = 0; b < B; ++b) {
        const float* __restrict__ lr = logits + b * T;
        float m = -3.402823466e38f;
        for (int i = tid; i < T; i += 1024) m = fmaxf(m, lr[i]);
        red[tid] = m;
        __syncthreads();
        for (int off = 512; off > 0; off >>= 1) {
            if (tid < off) red[tid] = fmaxf(red[tid], red[tid + off]);
            __syncthreads();
        }
        const float rmax = red[0];
        __syncthreads();
        float s = 0.0f;
        for (int i = tid; i < T; i += 1024) s += __expf(lr[i] - rmax);
        red[tid] = s;
        __syncthreads();
        for (int off = 512; off > 0; off >>= 1) {
            if (tid < off) red[tid] += red[tid + off];
            __syncthreads();
        }
        const float z = rmax + __logf(red[0]);
        __syncthreads();
        zsq += z * z;   // identical in every thread
    }

    if (tid == 0) {
        out[OUT_AUX] = AUX_W * bce / (float)(B * T);
        out[OUT_Z]   = Z_W * zsq / (float)B;
    }
}

extern "C" void kernel_launch(void* const* d_in, const int* in_sizes, int n_in,
                              void* d_out, int out_size, void* d_ws, size_t ws_size,
                              hipStream_t stream)
{
    const float* x = (const float*)d_in[0];   // (B, T, D) f32
    const float* W = (const float*)d_in[1];   // (1, D)    f32
    float* out = (float*)d_out;
    float* ws  = (float*)d_ws;
    float* gv  = ws + WS_GV;
    float* tgt = ws + WS_TGT;

    // 1) logits + gate values: one streaming pass over x (HBM-bound), WMMA f32.
    router_logits_wmma<<<(B * T) / 128, 256, 0, stream>>>(x, W, out, gv);
    // 2) per-row top-k select (sorted ascending) + gate weights + target mask.
    topk_select<<<B, 1024, 0, stream>>>(gv, out, tgt);
    // 3) aux BCE + z-loss scalars.
    losses_kernel<<<1, 1024, 0, stream>>>(out, tgt);

    (void)in_sizes; (void)n_in; (void)out_size; (void)ws_size;
}